// RTDETR_postprocess_76656576299739
// MI455X (gfx1250) — compile-verified
//
#include <hip/hip_runtime.h>
#include <hip/hip_bf16.h>

// RT-DETR post-process NMS for MI455X (gfx1250, wave32).
//   K1: single-WG LDS bitonic sort of 8192 (score,idx) u64 keys (64KB LDS).
//   K2: 16x16-tile suppression-bit matrix; TDM tensor_load_to_lds stages box
//       tiles, V_WMMA_F32_16X16X4_F32 computes the pairwise area-sum (outer
//       product [area,1]x[1;area]) while VALU does the min/max intersection.
//   K3: single-WG serial greedy scan over the bitmask + output emit.
// ws layout: boxes[8192*4] f32 | scores[8192] f32 | mask[8192*256] u32 (~8.6MB)

#define NBOX    8192
#define NWORDS  (NBOX / 32)   // 256 u32 mask words per row
#define NTILE16 (NBOX / 16)   // 512 16-wide tiles
#define SCORE_TH 0.5f
#define IOU_TH   0.5f

typedef float v2f  __attribute__((ext_vector_type(2)));
typedef float v8f  __attribute__((ext_vector_type(8)));
typedef unsigned int uint4v __attribute__((ext_vector_type(4)));
typedef int int8v __attribute__((ext_vector_type(8)));
typedef int int4v __attribute__((ext_vector_type(4)));

#if defined(__has_builtin)
#if __has_builtin(__builtin_amdgcn_tensor_load_to_lds)
#define HAVE_TDM 1
#endif
#if __has_builtin(__builtin_amdgcn_wmma_f32_16x16x4_f32)
#define HAVE_WMMA 1
#endif
#endif
#ifndef HAVE_TDM
#define HAVE_TDM 0
#endif
#ifndef HAVE_WMMA
#define HAVE_WMMA 0
#endif

// ---------------------------------------------------------------------------
// Kernel 1: max-reduce conf, normalize, bitonic sort descending in LDS.
// ---------------------------------------------------------------------------
__global__ __launch_bounds__(1024)
void rtdetr_prep_sort(const float* __restrict__ raw,
                      float* __restrict__ boxes,
                      float* __restrict__ scores) {
  __shared__ unsigned long long skey[NBOX];   // 64KB; aliased as float scratch
  float* f = (float*)skey;
  const unsigned t = threadIdx.x;

  // max(conf): conf is uniform[0,1) so 0 is a safe identity.
  float m = 0.0f;
  for (unsigned e = t; e < NBOX; e += 1024) m = fmaxf(m, raw[e * 5 + 4]);
  f[t] = m;
  __syncthreads();
  for (unsigned s = 512; s > 0; s >>= 1) {
    if (t < s) f[t] = fmaxf(f[t], f[t + s]);
    __syncthreads();
  }
  const float mx = f[0];
  __syncthreads();

  // key = (bits(conf/mx) << 32) | (N-1-idx): descending u64 sort gives
  // descending score with stable (ascending-index) tie order.
  for (unsigned e = t; e < NBOX; e += 1024) {
    float c = raw[e * 5 + 4] / mx;
    skey[e] = ((unsigned long long)__float_as_uint(c) << 32) |
              (unsigned long long)(NBOX - 1u - e);
  }
  __syncthreads();

  for (unsigned k = 2; k <= NBOX; k <<= 1) {
    for (unsigned j = k >> 1; j > 0; j >>= 1) {
      for (unsigned i = t; i < NBOX; i += 1024) {
        unsigned ixj = i ^ j;
        if (ixj > i) {
          unsigned long long a = skey[i], b = skey[ixj];
          bool up = ((i & k) == 0);              // descending network
          if (up ? (a < b) : (a > b)) { skey[i] = b; skey[ixj] = a; }
        }
      }
      __syncthreads();
    }
  }

  for (unsigned s = t; s < NBOX; s += 1024) {
    unsigned long long key = skey[s];
    float sc = __uint_as_float((unsigned)(key >> 32));
    unsigned idx = (NBOX - 1u) - (unsigned)(key & 0xFFFFFFFFull);
    float cx = raw[idx * 5 + 0], cy = raw[idx * 5 + 1];
    float w  = raw[idx * 5 + 2], h  = raw[idx * 5 + 3];
    boxes[s * 4 + 0] = cx - 0.5f * w;
    boxes[s * 4 + 1] = cy - 0.5f * h;
    boxes[s * 4 + 2] = cx + 0.5f * w;
    boxes[s * 4 + 3] = cy + 0.5f * h;
    scores[s] = sc;
  }
}

// ---------------------------------------------------------------------------
// TDM: 1-D DMA of nElems f32 from global to LDS (one transfer per wave).
// D# group0: count=1 | lds_addr | global_addr | type=2.
// D# group1: data_size=4B, tensor_dim0=nElems, tensor_dim1=1, tile_dim0=nElems.
// clang-23 toolchain: 6-arg builtin (g0 v4u, g1 v8i, g2 v4i, g3 v4i, v8i, cpol).
// ---------------------------------------------------------------------------
__device__ __forceinline__
void tdm_load_f32_1d(unsigned ldsAddr, const float* gsrc, unsigned nElems) {
#if HAVE_TDM
  unsigned long long ga = (unsigned long long)(const void*)gsrc;
  uint4v g0;
  g0[0] = 1u;                                               // count = 1 (valid)
  g0[1] = ldsAddr;                                          // lds_addr (bytes)
  g0[2] = (unsigned)ga;                                     // global_addr[31:0]
  g0[3] = (unsigned)((ga >> 32) & 0x1FFFFFFull) | (2u << 30); // hi addr | type=2
  int8v g1;
  g1[0] = (int)(2u << 16);                                  // data_size = 4B
  g1[1] = (int)((nElems & 0xFFFFu) << 16);                  // tensor_dim0 lo16
  g1[2] = (int)(((nElems >> 16) & 0xFFFFu) | (1u << 16));   // dim0 hi | dim1=1
  g1[3] = (int)((nElems & 0xFFFFu) << 16);                  // tile_dim0
  g1[4] = 0;                                                // tile_dim1/2 unused
  g1[5] = (int)nElems;                                      // dim0_stride lo32
  g1[6] = 0;
  g1[7] = 0;
  int4v gz4 = (int4v){0, 0, 0, 0};
  int8v gz8 = (int8v){0, 0, 0, 0, 0, 0, 0, 0};
  __builtin_amdgcn_tensor_load_to_lds(g0, g1, gz4, gz4, gz8, 0);
#else
  (void)ldsAddr; (void)gsrc; (void)nElems;
#endif
}

// ---------------------------------------------------------------------------
// Kernel 2: suppression bitmask. block = (32,8): 8 waves, each owns one
// 16-row x 16-col tile pair. mask16[row][colTile] halfword, bit b = col
// colTile*16+b suppressed-by-row (iou>0.5 && col>row).
// ---------------------------------------------------------------------------
__global__ __launch_bounds__(256)
void rtdetr_iou_mask(const float* __restrict__ boxes,
                     unsigned short* __restrict__ mask16) {
  __shared__ float sRow[64];        // 16 row boxes (xyxy)
  __shared__ float sCol[8][64];     // 16 col boxes per wave
  const unsigned lane = threadIdx.x;
  const unsigned wv = threadIdx.y;
  const unsigned rowTile = blockIdx.y;
  const unsigned colTile = blockIdx.x * 8u + wv;

#if HAVE_TDM
  if (wv == 0)
    tdm_load_f32_1d((unsigned)(unsigned long long)(void*)&sRow[0],
                    boxes + rowTile * 64u, 64u);
  tdm_load_f32_1d((unsigned)(unsigned long long)(void*)&sCol[wv][0],
                  boxes + colTile * 64u, 64u);
  __builtin_amdgcn_s_wait_tensorcnt(0);
#else
  if (wv == 0) {
    for (unsigned e = lane; e < 64u; e += 32u) sRow[e] = boxes[rowTile * 64u + e];
  }
  for (unsigned e = lane; e < 64u; e += 32u) sCol[wv][e] = boxes[colTile * 64u + e];
#endif
  __syncthreads();

  const unsigned n = lane & 15u;     // D-matrix: lane -> column n
  const unsigned grp = lane >> 4;    // lanes 16-31 hold rows 8..15
  const float cx1 = sCol[wv][n * 4 + 0], cy1 = sCol[wv][n * 4 + 1];
  const float cx2 = sCol[wv][n * 4 + 2], cy2 = sCol[wv][n * 4 + 3];
  const float colArea = (cx2 - cx1) * (cy2 - cy1);

  v8f d;
#if HAVE_WMMA
  // area_i + area_j as a 16x16x4 f32 WMMA: A = [area_row, 1 | 0 0],
  // B = [ones ; area_col ; 0 ; 0].  A layout: lanes 0-15 = (K0,K1), 16-31 = (K2,K3).
  // B layout: v0 = {K0 row | K1 row}, v1 = {K2 | K3}.
  v2f a, b;
  if (lane < 16u) {
    float rx1 = sRow[lane * 4 + 0], ry1 = sRow[lane * 4 + 1];
    float rx2 = sRow[lane * 4 + 2], ry2 = sRow[lane * 4 + 3];
    a.x = (rx2 - rx1) * (ry2 - ry1);  // A[m][0] = area_row(m)
    a.y = 1.0f;                       // A[m][1] = 1
    b.x = 1.0f;                       // B[0][n] = 1
  } else {
    a.x = 0.0f; a.y = 0.0f;           // K=2,3 unused
    b.x = colArea;                    // B[1][n] = area_col(n)
  }
  b.y = 0.0f;
  v8f cacc = (v8f){0, 0, 0, 0, 0, 0, 0, 0};
  d = __builtin_amdgcn_wmma_f32_16x16x4_f32(false, a, false, b, (short)0, cacc,
                                            false, false);
#else
#pragma unroll
  for (int r = 0; r < 8; ++r) {
    unsigned mrow = grp * 8u + (unsigned)r;
    float rx1 = sRow[mrow * 4 + 0], ry1 = sRow[mrow * 4 + 1];
    float rx2 = sRow[mrow * 4 + 2], ry2 = sRow[mrow * 4 + 3];
    d[r] = (rx2 - rx1) * (ry2 - ry1) + colArea;
  }
#endif

#pragma unroll
  for (int r = 0; r < 8; ++r) {
    unsigned mrow = grp * 8u + (unsigned)r;            // D VGPR r -> row mrow
    float rx1 = sRow[mrow * 4 + 0], ry1 = sRow[mrow * 4 + 1];
    float rx2 = sRow[mrow * 4 + 2], ry2 = sRow[mrow * 4 + 3];
    float iw = fmaxf(fminf(rx2, cx2) - fmaxf(rx1, cx1), 0.0f);
    float ih = fmaxf(fminf(ry2, cy2) - fmaxf(ry1, cy1), 0.0f);
    float inter = iw * ih;
    float uni = d[r] - inter;                          // area_i+area_j - inter
    unsigned gi = rowTile * 16u + mrow;
    unsigned gj = colTile * 16u + n;
    // iou > 0.5  <=>  inter > 0.5*max(union, 1e-9); plus upper-triangle j>i.
    bool bit = (inter > IOU_TH * fmaxf(uni, 1e-9f)) && (gj > gi);
    unsigned bal = __builtin_amdgcn_ballot_w32(bit);
    // ballot bits[15:0] = row (tile,r) cols 0-15; bits[31:16] = row (tile,8+r).
    if (lane == 2u * (unsigned)r)
      mask16[(rowTile * 16u + (unsigned)r) * NTILE16 + colTile] =
          (unsigned short)(bal & 0xFFFFu);
    if (lane == 2u * (unsigned)r + 1u)
      mask16[(rowTile * 16u + 8u + (unsigned)r) * NTILE16 + colTile] =
          (unsigned short)(bal >> 16);
  }
}

// ---------------------------------------------------------------------------
// Kernel 3: serial greedy scan over keep bits (LDS) + output emit.
// ---------------------------------------------------------------------------
__global__ __launch_bounds__(256)
void rtdetr_nms_scan(const float* __restrict__ boxes,
                     const float* __restrict__ scores,
                     const unsigned* __restrict__ mask,
                     float* __restrict__ out) {
  __shared__ unsigned keepw[NWORDS];
  volatile unsigned* vkeep = keepw;
  const unsigned t = threadIdx.x;

  unsigned w = 0u;
  for (unsigned b = 0; b < 32u; ++b)
    w |= (scores[t * 32u + b] >= SCORE_TH ? 1u : 0u) << b;
  keepw[t] = w;
  __syncthreads();

  for (unsigned i = 0; i < NBOX; ++i) {
    unsigned wi = vkeep[i >> 5];
    bool alive = (wi >> (i & 31u)) & 1u;   // uniform across block
    __syncthreads();                        // all reads before any writes
    if (alive) vkeep[t] &= ~mask[i * NWORDS + t];
    __syncthreads();
  }

  for (unsigned j = t; j < NBOX; j += 256u) {
    bool k = (keepw[j >> 5] >> (j & 31u)) & 1u;
    float x1 = boxes[j * 4 + 0], y1 = boxes[j * 4 + 1];
    float x2 = boxes[j * 4 + 2], y2 = boxes[j * 4 + 3];
    float sc = scores[j];
    out[j * 5 + 0] = k ? x1 : 0.0f;
    out[j * 5 + 1] = k ? y1 : 0.0f;
    out[j * 5 + 2] = k ? x2 : 0.0f;
    out[j * 5 + 3] = k ? y2 : 0.0f;
    out[j * 5 + 4] = k ? sc : 0.0f;
  }
}

extern "C" void kernel_launch(void* const* d_in, const int* in_sizes, int n_in,
                              void* d_out, int out_size, void* d_ws, size_t ws_size,
                              hipStream_t stream) {
  (void)in_sizes; (void)n_in; (void)out_size; (void)ws_size;
  const float* raw = (const float*)d_in[0];
  float* out = (float*)d_out;

  float* boxes = (float*)d_ws;              // 8192*4 f32
  float* scores = boxes + NBOX * 4;         // 8192 f32
  unsigned* mask = (unsigned*)(scores + NBOX);  // 8192*256 u32 (8MB)

  rtdetr_prep_sort<<<1, 1024, 0, stream>>>(raw, boxes, scores);
  rtdetr_iou_mask<<<dim3(NTILE16 / 8, NTILE16), dim3(32, 8), 0, stream>>>(
      boxes, (unsigned short*)mask);
  rtdetr_nms_scan<<<1, 256, 0, stream>>>(boxes, scores, mask, out);
}